// QuaternionMisorientation3Dqu_81372450390681
// MI455X (gfx1250) — compile-verified
//
#include <hip/hip_runtime.h>
#include <hip/hip_bf16.h>

typedef float v2f __attribute__((ext_vector_type(2)));
typedef float v8f __attribute__((ext_vector_type(8)));

#define C7 0.70710678118654752f
// 24 cubic symmetry quaternions (w,x,y,z), padded to 32 rows with zeros so the
// second WMMA A-tile (rows 16..31) needs no divergent bounds check.
__device__ __constant__ float c_S[32][4] = {
    { 1.f, 0.f, 0.f, 0.f}, { 0.f, 1.f, 0.f, 0.f}, { 0.f, 0.f, 1.f, 0.f}, { 0.f, 0.f, 0.f, 1.f},
    { 0.f, 0.f,  C7,  C7}, { 0.f, 0.f,  C7, -C7}, { 0.f,  C7, 0.f,  C7}, { 0.f,  C7, 0.f, -C7},
    { 0.f,  C7, -C7, 0.f}, { 0.f, -C7, -C7, 0.f},
    { 0.5f, 0.5f, 0.5f, 0.5f}, {-0.5f, 0.5f, 0.5f, 0.5f}, {-0.5f, 0.5f, 0.5f,-0.5f}, {-0.5f, 0.5f,-0.5f, 0.5f},
    {-0.5f,-0.5f, 0.5f, 0.5f}, {-0.5f,-0.5f, 0.5f,-0.5f}, {-0.5f,-0.5f,-0.5f, 0.5f}, {-0.5f, 0.5f,-0.5f,-0.5f},
    {-C7, 0.f, 0.f,  C7}, { C7, 0.f, 0.f,  C7}, {-C7, 0.f,  C7, 0.f}, {-C7, 0.f, -C7, 0.f},
    {-C7,  C7, 0.f, 0.f}, {-C7, -C7, 0.f, 0.f},
    {0,0,0,0},{0,0,0,0},{0,0,0,0},{0,0,0,0},{0,0,0,0},{0,0,0,0},{0,0,0,0},{0,0,0,0}
};

#define SPATIAL   (64*64*64)        // 262144 = 1<<18
#define N_ELEM    (8*SPATIAL)       // 2097152
#define N_GROUPS  (N_ELEM/32)       // 65536 groups of 32 elements
#define THREADS   256
#define WPB       (THREADS/32)      // waves per block
#define BLOCKS    2048              // 16384 waves -> 4 groups per wave

__global__ __launch_bounds__(THREADS)
void qmis_main(const float* __restrict__ x, const float* __restrict__ xh,
               float* __restrict__ partial) {
    const int lane  = threadIdx.x & 31;
    const int wave  = threadIdx.x >> 5;
    const int gwave = blockIdx.x * WPB + wave;
    const int nwav  = gridDim.x * WPB;
    const int col   = lane & 15;
    const bool lo   = lane < 16;

    // A operand layout (16x4 f32): lanes 0-15 hold (K0,K1) of row M=lane,
    // lanes 16-31 hold (K2,K3) of row M=lane-16.  Two tiles: S rows 0-15, 16-31(pad).
    const int k0 = lo ? 0 : 2, k1 = lo ? 1 : 3;
    v2f a0, a1;
    a0.x = c_S[col][k0];      a0.y = c_S[col][k1];
    a1.x = c_S[col + 16][k0]; a1.y = c_S[col + 16][k1];

    float acc = 0.f;
    for (int grp = gwave; grp < N_GROUPS; grp += nwav) {
        // 32 distinct elements per wave iteration; (b, spatial) remap keeps
        // every component stream a contiguous 128B run across the wave.
        const int elem = grp * 32 + lane;
        const int b    = elem >> 18;
        const int sp   = elem & (SPATIAL - 1);
        const size_t base = ((size_t)b << 20) + (size_t)sp; // (b*4)*SPATIAL + sp

        const float q0 = x[base];
        const float q1 = x[base + (size_t)(1 << 18)];
        const float q2 = x[base + (size_t)(2 << 18)];
        const float q3 = x[base + (size_t)(3 << 18)];
        const float h0 = xh[base];
        const float h1 = xh[base + (size_t)(1 << 18)];
        const float h2 = xh[base + (size_t)(2 << 18)];
        const float h3 = xh[base + (size_t)(3 << 18)];

        // r = q_hat (x) conj(q) : dots[n] = <S_n, r>
        const float r0 =  h0*q0 + h1*q1 + h2*q2 + h3*q3;
        const float r1 = -h0*q1 + h1*q0 - h2*q3 + h3*q2;
        const float r2 = -h0*q2 + h1*q3 + h2*q0 - h3*q1;
        const float r3 = -h0*q3 - h1*q2 + h2*q1 + h3*q0;

        // Partner-half copies (lane XOR 16) to fill the opposite B slots.
        const float s0 = __shfl_xor(r0, 16, 32);
        const float s1 = __shfl_xor(r1, 16, 32);
        const float s2 = __shfl_xor(r2, 16, 32);
        const float s3 = __shfl_xor(r3, 16, 32);

        // B operand (4x16 f32): VGPR0 = rows K0 (lanes 0-15) / K2 (16-31),
        //                       VGPR1 = rows K1 / K3.
        v2f b0, b1;                 // element-tile 0 (elems 0-15), tile 1 (16-31)
        b0.x = lo ? r0 : s2;  b0.y = lo ? r1 : s3;
        b1.x = lo ? s0 : r2;  b1.y = lo ? s1 : r3;

        v8f cz = {};
        v8f d00 = __builtin_amdgcn_wmma_f32_16x16x4_f32(false, a0, false, b0,
                                                        (short)0, cz, false, false);
        v8f d01 = __builtin_amdgcn_wmma_f32_16x16x4_f32(false, a1, false, b0,
                                                        (short)0, cz, false, false);
        v8f d10 = __builtin_amdgcn_wmma_f32_16x16x4_f32(false, a0, false, b1,
                                                        (short)0, cz, false, false);
        v8f d11 = __builtin_amdgcn_wmma_f32_16x16x4_f32(false, a1, false, b1,
                                                        (short)0, cz, false, false);

        float m0 = 0.f, m1 = 0.f;
#pragma unroll
        for (int i = 0; i < 8; ++i) {
            m0 = fmaxf(m0, fmaxf(fabsf(d00[i]), fabsf(d01[i])));
            m1 = fmaxf(m1, fmaxf(fabsf(d10[i]), fabsf(d11[i])));
        }
        // combine sym rows held in the partner half-wave (same element column)
        m0 = fmaxf(m0, __shfl_xor(m0, 16, 32));
        m1 = fmaxf(m1, __shfl_xor(m1, 16, 32));

        // each half-wave finishes its own 16 elements: one acos per element
        float m = lo ? m0 : m1;
        m = fminf(m, 1.0f);
        const float t = acosf(m);
        acc += t * t;
    }

    // wave reduction
#pragma unroll
    for (int o = 16; o > 0; o >>= 1) acc += __shfl_xor(acc, o, 32);

    __shared__ float smem[WPB];
    if (lane == 0) smem[wave] = acc;
    __syncthreads();
    if (threadIdx.x == 0) {
        float s = 0.f;
#pragma unroll
        for (int i = 0; i < WPB; ++i) s += smem[i];
        partial[blockIdx.x] = s;
    }
}

__global__ __launch_bounds__(256)
void qmis_reduce(const float* __restrict__ partial, float* __restrict__ out, int n) {
    __shared__ float smem[256];
    float s = 0.f;
    for (int i = threadIdx.x; i < n; i += 256) s += partial[i];
    smem[threadIdx.x] = s;
    __syncthreads();
    for (int stride = 128; stride > 0; stride >>= 1) {
        if (threadIdx.x < stride) smem[threadIdx.x] += smem[threadIdx.x + stride];
        __syncthreads();
    }
    if (threadIdx.x == 0) out[0] = smem[0] * (1.0f / (float)N_ELEM);
}

extern "C" void kernel_launch(void* const* d_in, const int* in_sizes, int n_in,
                              void* d_out, int out_size, void* d_ws, size_t ws_size,
                              hipStream_t stream) {
    const float* x  = (const float*)d_in[0];
    const float* xh = (const float*)d_in[1];
    float* out      = (float*)d_out;
    float* partial  = (float*)d_ws;   // BLOCKS floats

    qmis_main<<<BLOCKS, THREADS, 0, stream>>>(x, xh, partial);
    qmis_reduce<<<1, 256, 0, stream>>>(partial, out, BLOCKS);
}